// HISCLoss_15857019257334
// MI455X (gfx1250) — compile-verified
//
#include <hip/hip_runtime.h>

// HSIC loss, algebraically reduced:
//   trace(center(XX^T) * center(YY^T)) = || X^T Y - B * mx my^T ||_F^2
// so each of the 6 expert pairs is a 256x256 (d x d) fp32 GEMM with K=4096
// (V_WMMA_F32_16X16X4_F32) plus a rank-1 mean correction, then a squared-
// Frobenius reduction. K is staged through double-buffered LDS slabs filled
// with GLOBAL_LOAD_ASYNC_TO_LDS_B32 (per-lane LDS scatter does the
// row-major -> K-contiguous transpose for free, tracked by ASYNCcnt).

#define B_ROWS  4096
#define D_COLS  256
#define NPAIRS  6
#define KSLAB   32
#define LPAD    34                  // even pitch -> 8B-aligned v2f, conflict-free banks
#define NSTAGES (B_ROWS / KSLAB)

typedef __attribute__((ext_vector_type(2))) float v2f;
typedef __attribute__((ext_vector_type(8))) float v8f;

#if defined(__has_builtin)
#if __has_builtin(__builtin_amdgcn_global_load_async_to_lds_b32)
#define HAVE_ASYNC_LDS 1
#endif
#endif
#ifndef HAVE_ASYNC_LDS
#define HAVE_ASYNC_LDS 0
#endif

// Builtin signature (from clang diagnostic): global side is `int addrspace(1)*`
// (prints as `__device__ int*`), LDS side correspondingly `int addrspace(3)*`.
// AS1 pointer = 64-bit VA; AS3 pointer = 32-bit LDS offset = low 32 bits of
// the generic LDS address. Pure inttoptr casts sidestep addrspacecast/const
// legality questions.
#define GPTR(p) ((__attribute__((address_space(1))) int*)(uintptr_t)(p))
#define LPTR(p) ((__attribute__((address_space(3))) int*)(unsigned)(uintptr_t)(p))

__device__ __forceinline__ void wait_async_zero() {
#if HAVE_ASYNC_LDS
#if __has_builtin(__builtin_amdgcn_s_wait_asynccnt)
    __builtin_amdgcn_s_wait_asynccnt(0);
#else
    asm volatile("s_wait_asynccnt 0x0" ::: "memory");
#endif
#endif
}

__device__ __forceinline__ const float* pick(int i, const float* a, const float* b,
                                             const float* c, const float* d) {
    return i == 0 ? a : (i == 1 ? b : (i == 2 ? c : d));
}

// Fill one KSLAB x 64 slab of Ei (cols m0..m0+63) and Ej (cols n0..n0+63)
// into K-contiguous LDS arrays Ab/Bb (layout [c * LPAD + k]).
__device__ __forceinline__ void stage_slab(const float* __restrict__ Ei,
                                           const float* __restrict__ Ej,
                                           int ks, int m0, int n0, int tid,
                                           float* Ab, float* Bb) {
#if HAVE_ASYNC_LDS
    #pragma unroll
    for (int it = 0; it < 16; ++it) {
        int f = it * 128 + tid;        // 0..2047
        int k = f >> 6;                // 0..31
        int c = f & 63;                // 0..63
        __builtin_amdgcn_global_load_async_to_lds_b32(
            GPTR(&Ei[(size_t)(ks + k) * D_COLS + m0 + c]),
            LPTR(&Ab[c * LPAD + k]), 0, 0);
        __builtin_amdgcn_global_load_async_to_lds_b32(
            GPTR(&Ej[(size_t)(ks + k) * D_COLS + n0 + c]),
            LPTR(&Bb[c * LPAD + k]), 0, 0);
    }
#else
    // Fallback: batch all global loads (b128) into registers first so they
    // stay in flight together, then scatter-transpose into LDS.
    float4 ra[4], rb[4];
    int k0 = tid >> 4;                 // 0..7
    int c0 = (tid & 15) * 4;           // 0,4,...,60
    #pragma unroll
    for (int it = 0; it < 4; ++it) {
        int k = it * 8 + k0;
        ra[it] = *(const float4*)&Ei[(size_t)(ks + k) * D_COLS + m0 + c0];
        rb[it] = *(const float4*)&Ej[(size_t)(ks + k) * D_COLS + n0 + c0];
    }
    #pragma unroll
    for (int it = 0; it < 4; ++it) {
        int k = it * 8 + k0;
        Ab[(c0 + 0) * LPAD + k] = ra[it].x;
        Ab[(c0 + 1) * LPAD + k] = ra[it].y;
        Ab[(c0 + 2) * LPAD + k] = ra[it].z;
        Ab[(c0 + 3) * LPAD + k] = ra[it].w;
        Bb[(c0 + 0) * LPAD + k] = rb[it].x;
        Bb[(c0 + 1) * LPAD + k] = rb[it].y;
        Bb[(c0 + 2) * LPAD + k] = rb[it].z;
        Bb[(c0 + 3) * LPAD + k] = rb[it].w;
    }
#endif
}

// ---------------------------------------------------------------------------
// Kernel 1: column means for all 4 experts. 4 blocks x 1024 threads.
// ---------------------------------------------------------------------------
__global__ void col_mean_kernel(const float* __restrict__ e0, const float* __restrict__ e1,
                                const float* __restrict__ e2, const float* __restrict__ e3,
                                float* __restrict__ means) {
    const float* src = pick(blockIdx.x, e0, e1, e2, e3);
    int t    = threadIdx.x;
    int col  = t & (D_COLS - 1);
    int part = t >> 8;                 // 0..3
    float s = 0.0f;
    int r0 = part * (B_ROWS / 4);
    for (int r = r0; r < r0 + (B_ROWS / 4); ++r)
        s += src[(size_t)r * D_COLS + col];
    __shared__ float red[1024];
    red[t] = s;
    __syncthreads();
    if (t < D_COLS) {
        float m = (red[t] + red[t + 256] + red[t + 512] + red[t + 768]) * (1.0f / B_ROWS);
        means[blockIdx.x * D_COLS + col] = m;
    }
}

// ---------------------------------------------------------------------------
// Kernel 2: per-pair d x d GEMM via WMMA f32 16x16x4 + centered reduction.
// Grid: 6 pairs * 16 tiles (64x64 output each). Block: 128 threads = 4 waves,
// each wave owns a 32x32 block (2x2 accumulators, 2x A/B fragment reuse).
// Double-buffered LDS slabs, async-filled; one barrier per stage.
// ---------------------------------------------------------------------------
__global__ void hisc_pair_kernel(const float* __restrict__ e0, const float* __restrict__ e1,
                                 const float* __restrict__ e2, const float* __restrict__ e3,
                                 const float* __restrict__ means,
                                 float* __restrict__ partials) {
    __shared__ float As[2][64 * LPAD];   // [buf][c*LPAD + k]
    __shared__ float Bs[2][64 * LPAD];
    __shared__ float wred[4];

    const int PI[NPAIRS] = {0, 0, 0, 1, 1, 2};
    const int PJ[NPAIRS] = {1, 2, 3, 2, 3, 3};

    int pair = blockIdx.x >> 4;        // 0..5
    int tile = blockIdx.x & 15;        // 0..15
    int m0   = (tile >> 2) * 64;       // Ei column block (A = Ei^T row block)
    int n0   = (tile & 3) * 64;        // Ej column block

    const float* Ei = pick(PI[pair], e0, e1, e2, e3);
    const float* Ej = pick(PJ[pair], e0, e1, e2, e3);

    int tid  = threadIdx.x;
    int lane = tid & 31;
    int wave = tid >> 5;               // 0..3
    int wm   = (wave >> 1) * 32;       // wave's M offset in 64x64 tile
    int wn   = (wave & 1) * 32;        // wave's N offset
    int l16  = lane & 15;
    int koff = (lane >> 4) << 1;       // lanes 0-15 -> K {0,1}; 16-31 -> K {2,3}

    v8f acc00 = {}, acc01 = {}, acc10 = {}, acc11 = {};

    stage_slab(Ei, Ej, 0, m0, n0, tid, As[0], Bs[0]);

    for (int s = 0; s < NSTAGES; ++s) {
        int cur = s & 1;
        wait_async_zero();             // own slab-fill complete (LDS written)
        __syncthreads();               // everyone's fill complete / prev compute done
        if (s + 1 < NSTAGES)           // overlap next fill with this compute
            stage_slab(Ei, Ej, (s + 1) * KSLAB, m0, n0, tid, As[cur ^ 1], Bs[cur ^ 1]);

        const float* Ab = As[cur];
        const float* Bb = Bs[cur];
        #pragma unroll
        for (int kk = 0; kk < KSLAB; kk += 4) {
            int kb = kk + koff;        // even -> 8B-aligned v2f LDS loads
            v2f a0 = *(const v2f*)&Ab[(wm      + l16) * LPAD + kb];
            v2f a1 = *(const v2f*)&Ab[(wm + 16 + l16) * LPAD + kb];
            v2f b0 = *(const v2f*)&Bb[(wn      + l16) * LPAD + kb];
            v2f b1 = *(const v2f*)&Bb[(wn + 16 + l16) * LPAD + kb];
            acc00 = __builtin_amdgcn_wmma_f32_16x16x4_f32(false, a0, false, b0,
                                                          (short)0, acc00, false, false);
            acc01 = __builtin_amdgcn_wmma_f32_16x16x4_f32(false, a0, false, b1,
                                                          (short)0, acc01, false, false);
            acc10 = __builtin_amdgcn_wmma_f32_16x16x4_f32(false, a1, false, b0,
                                                          (short)0, acc10, false, false);
            acc11 = __builtin_amdgcn_wmma_f32_16x16x4_f32(false, a1, false, b1,
                                                          (short)0, acc11, false, false);
        }
    }

    // Rank-1 mean correction + squared-Frobenius reduction.
    // C/D layout: VGPR r, lane L -> (M = r + 8*(L>=16), N = L%16).
    const float* mi = means + PI[pair] * D_COLS + m0 + wm;
    const float* mj = means + PJ[pair] * D_COLS + n0 + wn;
    int rowOff = (lane >> 4) * 8;
    float mj0 = mj[l16];
    float mj1 = mj[16 + l16];
    float ssum = 0.0f;
    #pragma unroll
    for (int r = 0; r < 8; ++r) {
        float mi0 = mi[r + rowOff];
        float mi1 = mi[16 + r + rowOff];
        float c;
        c = acc00[r] - (float)B_ROWS * mi0 * mj0; ssum += c * c;
        c = acc01[r] - (float)B_ROWS * mi0 * mj1; ssum += c * c;
        c = acc10[r] - (float)B_ROWS * mi1 * mj0; ssum += c * c;
        c = acc11[r] - (float)B_ROWS * mi1 * mj1; ssum += c * c;
    }
    #pragma unroll
    for (int off = 16; off > 0; off >>= 1)
        ssum += __shfl_xor(ssum, off, 32);
    if (lane == 0) wred[wave] = ssum;
    __syncthreads();
    if (tid == 0)
        partials[blockIdx.x] = wred[0] + wred[1] + wred[2] + wred[3];
}

// ---------------------------------------------------------------------------
// Kernel 3: fold 96 block partials -> scalar loss.
// loss = 0.1 * mean_over_6_pairs( sumsq / (B-1)^2 )
// ---------------------------------------------------------------------------
__global__ void finalize_kernel(const float* __restrict__ partials, float* __restrict__ out) {
    __shared__ float r[4];
    int t = threadIdx.x;               // 128 threads
    float s = (t < NPAIRS * 16) ? partials[t] : 0.0f;
    #pragma unroll
    for (int off = 16; off > 0; off >>= 1)
        s += __shfl_xor(s, off, 32);
    if ((t & 31) == 0) r[t >> 5] = s;
    __syncthreads();
    if (t == 0) {
        float tot = r[0] + r[1] + r[2] + r[3];
        const float scale = (float)(0.1 / (6.0 * 4095.0 * 4095.0));
        out[0] = tot * scale;
    }
}

extern "C" void kernel_launch(void* const* d_in, const int* in_sizes, int n_in,
                              void* d_out, int out_size, void* d_ws, size_t ws_size,
                              hipStream_t stream) {
    const float* e0 = (const float*)d_in[0];
    const float* e1 = (const float*)d_in[1];
    const float* e2 = (const float*)d_in[2];
    const float* e3 = (const float*)d_in[3];
    float* means    = (float*)d_ws;            // 4*256 floats
    float* partials = means + 4 * D_COLS;      // 96 floats
    float* out      = (float*)d_out;

    col_mean_kernel<<<4, 1024, 0, stream>>>(e0, e1, e2, e3, means);
    hisc_pair_kernel<<<NPAIRS * 16, 128, 0, stream>>>(e0, e1, e2, e3, means, partials);
    finalize_kernel<<<1, 128, 0, stream>>>(partials, out);
}